// Mamba2Layer_81157702025682
// MI455X (gfx1250) — compile-verified
//
#include <hip/hip_runtime.h>
#include <hip/hip_bf16.h>
#include <math.h>

// ---------------- problem constants (from reference) ----------------
#define BB      2
#define TT      2048
#define DM      1024
#define DI      2048            // d_inner
#define DS      128             // d_state
#define DCONV   4
#define NH      32              // heads
#define HP      64              // headdim
#define DPROJ   4384            // 2*DI + 2*DS + NH
#define BT      (BB * TT)       // 4096 tokens
#define EPSV    1e-5f

// WMMA fragment types (wave32): f32 16x16x4 -> A/B = 2 floats/lane, C/D = 8 floats/lane
typedef __attribute__((ext_vector_type(2))) float v2f;
typedef __attribute__((ext_vector_type(8))) float v8f;

// =====================================================================
// GEMM (NT):  C[m,n] = sum_k A[m*lda + k] * Bw[n*ldb + k]
// One wave computes one 16x16 tile via V_WMMA_F32_16X16X4_F32.
// 4 waves / block stack along M (block covers 64 rows of M).
// Layouts per CDNA5 ISA 7.12.2:
//   A 16x4 : lane(l<16) row=M0+l holds K={0,1}; lane(l>=16) row=M0+l-16 holds K={2,3}
//   B 4x16 : mirrored with N in place of M
//   C/D    : vgpr r -> M = r + 8*(l>=16), N = l&15
// =====================================================================
__global__ __launch_bounds__(128)
void wmma_gemm_nt(const float* __restrict__ A, const float* __restrict__ Bw,
                  float* __restrict__ C,
                  int M, int N, int K, int lda, int ldb, int ldc) {
    const int wave = threadIdx.x >> 5;
    const int lane = threadIdx.x & 31;
    const int n0 = blockIdx.x * 16;
    const int m0 = (blockIdx.y * 4 + wave) * 16;
    if (m0 >= M || n0 >= N) return;   // wave-uniform: EXEC stays all-ones inside

    const int half = lane >> 4;       // 0: K {0,1} | 1: K {2,3}
    const int koff = half * 2;
    const float* __restrict__ pa = A  + (size_t)(m0 + (lane & 15)) * lda + koff;
    const float* __restrict__ pb = Bw + (size_t)(n0 + (lane & 15)) * ldb + koff;

    v8f acc = {0.f, 0.f, 0.f, 0.f, 0.f, 0.f, 0.f, 0.f};
#pragma unroll 4
    for (int k = 0; k < K; k += 4) {
        v2f a = *(const v2f*)(pa + k);
        v2f b = *(const v2f*)(pb + k);
        __builtin_prefetch(pb + k + 128, 0, 1);
        // D = A*B + C  (8-arg form: neg_a, A, neg_b, B, c_mod, C, reuse_a, reuse_b)
        acc = __builtin_amdgcn_wmma_f32_16x16x4_f32(false, a, false, b,
                                                    (short)0, acc, false, false);
    }

    const int cm = m0 + (half << 3);
    float* __restrict__ pc = C + (size_t)cm * ldc + n0 + (lane & 15);
#pragma unroll
    for (int r = 0; r < 8; ++r) pc[(size_t)r * ldc] = acc[r];
}

// =====================================================================
// Causal depthwise conv1d (K=4) + bias + SiLU over proj[:, 0:DI] -> xh
// =====================================================================
__device__ __forceinline__ float silu_f(float v) {
    return v / (1.f + __expf(-v));
}

__global__ __launch_bounds__(256)
void conv_silu_kernel(const float* __restrict__ proj,
                      const float* __restrict__ conv_w,
                      const float* __restrict__ conv_b,
                      float* __restrict__ xh) {
    int idx = blockIdx.x * 256 + threadIdx.x;          // over BT*DI
    if (idx >= BT * DI) return;
    const int c   = idx % DI;
    const int t   = (idx / DI) % TT;
    const int b   = idx / (DI * TT);
    const size_t rowb = (size_t)(b * TT) * DPROJ + c;  // column c of batch b, time 0
    float acc = conv_b[c];
#pragma unroll
    for (int k = 0; k < DCONV; ++k) {
        const int ts = t - (DCONV - 1) + k;
        if (ts >= 0)
            acc = fmaf(conv_w[c * DCONV + k], proj[rowb + (size_t)ts * DPROJ], acc);
    }
    xh[(size_t)idx] = silu_f(acc);
}

// =====================================================================
// dt = softplus(dt_in + dt_bias); ab = exp(-exp(A_log) * dt)
// =====================================================================
__global__ __launch_bounds__(256)
void dtab_kernel(const float* __restrict__ proj,
                 const float* __restrict__ dt_bias,
                 const float* __restrict__ A_log,
                 float* __restrict__ dt_arr,
                 float* __restrict__ ab_arr) {
    int idx = blockIdx.x * 256 + threadIdx.x;          // over BT*NH
    if (idx >= BT * NH) return;
    const int h   = idx % NH;
    const int row = idx / NH;                          // token index b*T+t
    float raw = proj[(size_t)row * DPROJ + (2 * DI + 2 * DS) + h] + dt_bias[h];
    float dt  = (raw > 20.f) ? raw : log1pf(__expf(raw));
    dt_arr[idx] = dt;
    ab_arr[idx] = __expf(-__expf(A_log[h]) * dt);
}

// =====================================================================
// Selective scan: one workgroup per (b,h). 256 threads.
// thread tid: p = tid&63, n-chunk nc = tid>>6 covers n in [nc*32, nc*32+32).
// Each thread holds 32 state elements in registers (8192 floats / WG).
// y written in-place over xh (x at time t is consumed at time t).
// =====================================================================
__global__ __launch_bounds__(256)
void scan_kernel(const float* __restrict__ proj,
                 float* __restrict__ xh,
                 const float* __restrict__ dt_arr,
                 const float* __restrict__ ab_arr,
                 const float* __restrict__ Dp) {
    const int b = blockIdx.x / NH;
    const int h = blockIdx.x % NH;
    const int tid = threadIdx.x;
    const int p  = tid & 63;
    const int nc = tid >> 6;                 // 0..3

    __shared__ float sB[DS];
    __shared__ float sC[DS];
    __shared__ float sX[HP];
    __shared__ float sPart[256];

    float S[32];
#pragma unroll
    for (int i = 0; i < 32; ++i) S[i] = 0.f;

    const float Dh = Dp[h];

    for (int t = 0; t < TT; ++t) {
        const size_t row = (size_t)(b * TT + t);
        const float* __restrict__ prow = proj + row * DPROJ;
        float* __restrict__ xrow = xh + row * DI + h * HP;

        if (tid < DS)       sB[tid]       = prow[2 * DI + tid];
        else                sC[tid - DS]  = prow[2 * DI + DS + (tid - DS)];
        if (tid < HP)       sX[tid]       = xrow[tid];
        const float dt_t = dt_arr[row * NH + h];
        const float ab   = ab_arr[row * NH + h];
        __syncthreads();

        const float xp = sX[p];
        float ypart = 0.f;
#pragma unroll
        for (int i = 0; i < 32; ++i) {
            const int n = nc * 32 + i;
            S[i] = fmaf(ab, S[i], (sB[n] * dt_t) * xp);
            ypart = fmaf(sC[n], S[i], ypart);
        }
        sPart[tid] = ypart;
        __syncthreads();

        if (nc == 0) {
            float y = sPart[p] + sPart[64 + p] + sPart[128 + p] + sPart[192 + p];
            xrow[p] = fmaf(Dh, xp, y);       // fused skip: y + Dp * x
        }
    }
}

// =====================================================================
// Gated RMSNorm: one workgroup per token row.
// reads y from xh, z from proj[:, DI:2*DI]; writes into proj[:, 0:DI]
// (x_ssm columns are dead after the conv) so out_proj reads lda=DPROJ.
// =====================================================================
__global__ __launch_bounds__(256)
void rmsnorm_gate_kernel(const float* __restrict__ xh,
                         float* __restrict__ proj,
                         const float* __restrict__ norm_w) {
    const int row = blockIdx.x;
    const int tid = threadIdx.x;
    const float* __restrict__ yrow = xh + (size_t)row * DI;
    float* __restrict__ prow = proj + (size_t)row * DPROJ;

    __shared__ float red[256];
    float partial = 0.f;
#pragma unroll
    for (int k = 0; k < DI / 256; ++k) {
        const float v = yrow[tid + k * 256];
        partial = fmaf(v, v, partial);
    }
    red[tid] = partial;
    __syncthreads();
#pragma unroll
    for (int s = 128; s > 0; s >>= 1) {
        if (tid < s) red[tid] += red[tid + s];
        __syncthreads();
    }
    const float scale = rsqrtf(red[0] * (1.f / (float)DI) + EPSV);
#pragma unroll
    for (int k = 0; k < DI / 256; ++k) {
        const int c = tid + k * 256;
        const float z = prow[DI + c];
        prow[c] = yrow[c] * scale * norm_w[c] * silu_f(z);
    }
}

// =====================================================================
extern "C" void kernel_launch(void* const* d_in, const int* in_sizes, int n_in,
                              void* d_out, int out_size, void* d_ws, size_t ws_size,
                              hipStream_t stream) {
    const float* x          = (const float*)d_in[0];
    const float* in_proj_w  = (const float*)d_in[1];
    const float* conv_w     = (const float*)d_in[2];
    const float* conv_b     = (const float*)d_in[3];
    const float* A_log      = (const float*)d_in[4];
    const float* Dp         = (const float*)d_in[5];
    const float* dt_bias    = (const float*)d_in[6];
    const float* norm_w     = (const float*)d_in[7];
    const float* out_proj_w = (const float*)d_in[8];
    float* out = (float*)d_out;

    // workspace carve-up
    float* proj   = (float*)d_ws;                          // BT x DPROJ
    float* xh     = proj   + (size_t)BT * DPROJ;           // BT x DI (conv out, then y)
    float* dt_arr = xh     + (size_t)BT * DI;              // BT x NH
    float* ab_arr = dt_arr + (size_t)BT * NH;              // BT x NH

    // 1) in_proj: proj = x @ in_proj_w^T   (M=4096, N=4384, K=1024)
    wmma_gemm_nt<<<dim3(DPROJ / 16, (BT / 16) / 4), dim3(128), 0, stream>>>(
        x, in_proj_w, proj, BT, DPROJ, DM, DM, DM, DPROJ);

    // 2) causal depthwise conv + SiLU
    conv_silu_kernel<<<(BT * DI) / 256, 256, 0, stream>>>(proj, conv_w, conv_b, xh);

    // 2b) dt / A_bar precompute
    dtab_kernel<<<(BT * NH) / 256, 256, 0, stream>>>(proj, dt_bias, A_log, dt_arr, ab_arr);

    // 3) selective scan (+ fused Dp*x skip), y overwrites xh
    scan_kernel<<<BB * NH, 256, 0, stream>>>(proj, xh, dt_arr, ab_arr, Dp);

    // 4) gated RMSNorm -> proj[:, 0:DI]
    rmsnorm_gate_kernel<<<BT, 256, 0, stream>>>(xh, proj, norm_w);

    // 5) out_proj: out = ynorm @ out_proj_w^T   (M=4096, N=1024, K=2048)
    wmma_gemm_nt<<<dim3(DM / 16, (BT / 16) / 4), dim3(128), 0, stream>>>(
        proj, out_proj_w, out, BT, DM, DI, DPROJ, DI, DM);
}